// LinearAttention_70617852280995
// MI455X (gfx1250) — compile-verified
//
#include <hip/hip_runtime.h>

typedef __attribute__((ext_vector_type(2))) float v2f;
typedef __attribute__((ext_vector_type(8))) float v8f;

#define EPSF 1e-6f
constexpr int BN = 8;     // batch
constexpr int CD = 512;   // channels
constexpr int NT = 4096;  // tokens = H*W

// fp32 WMMA: D(16x16) = A(16x4) * B(4x16) + C
// 8-arg pattern: (neg_a, A, neg_b, B, c_mod, C, reuse_a, reuse_b)
__device__ __forceinline__ v8f wmma_f32(v2f a, v2f b, v8f c) {
  return __builtin_amdgcn_wmma_f32_16x16x4_f32(false, a, false, b, (short)0, c,
                                               false, false);
}

// Bank-swizzled 32x512 tile index: lanes that differ only in m (A-operand
// pattern) hit distinct banks: bank = (c+m) & 63.
__device__ __forceinline__ int qidx(int m, int c) {
  return m * CD + ((c + m) & (CD - 1));
}

// ---------------------------------------------------------------------------
// Kernel 1: K = relu(t@Wk + bk), V = t@Wv + bv   (t[b,n,c] = x[b,c,n])
// grid (NT/32, BN), block 256. LDS-staged A tile, 16 WMMA sub-tiles per wave.
// ---------------------------------------------------------------------------
__global__ __launch_bounds__(256) void k_proj_kv(
    const float* __restrict__ x, const float* __restrict__ Wk,
    const float* __restrict__ bk, const float* __restrict__ Wv,
    const float* __restrict__ bv, float* __restrict__ Kout,
    float* __restrict__ Vout) {
  __shared__ float a_lds[32 * CD];  // 64 KB
  const int b = blockIdx.y;
  const int m0 = blockIdx.x * 32;
  const int tid = threadIdx.x;
  const int wave = tid >> 5, lane = tid & 31;
  const int l15 = lane & 15, lhalf = lane >> 4;
  const int koff = lhalf * 2;

  // phase 0: a_lds[m][c] = x[b][c][m0+m]  (coalesced along m)
  const float* xb = x + (size_t)b * CD * NT;
  for (int i = tid; i < 32 * CD; i += 256) {
    int m = i & 31, c = i >> 5;
    a_lds[qidx(m, c)] = xb[(size_t)c * NT + m0 + m];
  }
  __syncthreads();

  // phase 1: 2 mats (K,V) x 2 m-subtiles x 32 n-subtiles = 128 sub-tiles
  for (int s = wave; s < 128; s += 8) {
    const int sel = s & 1;  // 0 = K (relu), 1 = V
    const int t2 = s >> 1;
    const int mi = t2 & 1;
    const int ni = t2 >> 1;
    const int col = ni * 16 + l15;
    const float* W = sel ? Wv : Wk;
    v8f acc = {};
#pragma unroll 4
    for (int k0 = 0; k0 < CD; k0 += 4) {
      v2f a, bm;
      a.x = a_lds[qidx(mi * 16 + l15, k0 + koff)];
      a.y = a_lds[qidx(mi * 16 + l15, k0 + koff + 1)];
      bm.x = W[(size_t)(k0 + koff) * CD + col];
      bm.y = W[(size_t)(k0 + koff + 1) * CD + col];
      acc = wmma_f32(a, bm, acc);
    }
    const float bias = sel ? bv[col] : bk[col];
    float* Dst = sel ? Vout : Kout;
#pragma unroll
    for (int j = 0; j < 8; ++j) {
      const int m = m0 + mi * 16 + j + lhalf * 8;
      float v = acc[j] + bias;
      if (sel == 0) v = fmaxf(v, 0.0f);
      Dst[((size_t)b * NT + m) * CD + col] = v;
    }
  }
}

// ---------------------------------------------------------------------------
// Kernel 2: ksum[b][c] += sum over n-chunk of K[b][n][c]   (ksum pre-zeroed)
// grid (CD/256, NT/256, BN), block 256.
// ---------------------------------------------------------------------------
__global__ __launch_bounds__(256) void k_ksum(const float* __restrict__ Kin,
                                              float* __restrict__ ksum) {
  const int d = blockIdx.x * 256 + threadIdx.x;
  const int n0 = blockIdx.y * 256;
  const int b = blockIdx.z;
  const float* Kb = Kin + (size_t)b * NT * CD;
  float s = 0.0f;
  for (int n = 0; n < 256; ++n) s += Kb[(size_t)(n0 + n) * CD + d];
  atomicAdd(&ksum[b * CD + d], s);
}

// ---------------------------------------------------------------------------
// Kernel 3: kv[b] = K[b]^T @ V[b]   (512x512, K-dim = 4096)
// grid (CD/32, CD/64, BN), block 256: wave -> (mi = w>>2, ni = w&3) sub-tile.
// ---------------------------------------------------------------------------
__global__ __launch_bounds__(256) void k_kvmat(const float* __restrict__ Kin,
                                               const float* __restrict__ Vin,
                                               float* __restrict__ kvout) {
  const int b = blockIdx.z;
  const int tid = threadIdx.x;
  const int wave = tid >> 5, lane = tid & 31;
  const int l15 = lane & 15, lhalf = lane >> 4;
  const int koff = lhalf * 2;
  const int mi = wave >> 2, ni = wave & 3;
  const int crow = blockIdx.x * 32 + mi * 16 + l15;  // A: M index
  const int dcol = blockIdx.y * 64 + ni * 16 + l15;  // B/D: N index
  const float* Kb = Kin + (size_t)b * NT * CD;
  const float* Vb = Vin + (size_t)b * NT * CD;
  v8f acc = {};
#pragma unroll 4
  for (int n = 0; n < NT; n += 4) {
    v2f a, bm;
    a.x = Kb[(size_t)(n + koff) * CD + crow];      // A[M=c][K=n] = K[n][c]
    a.y = Kb[(size_t)(n + koff + 1) * CD + crow];
    bm.x = Vb[(size_t)(n + koff) * CD + dcol];     // B[K=n][N=d] = V[n][d]
    bm.y = Vb[(size_t)(n + koff + 1) * CD + dcol];
    acc = wmma_f32(a, bm, acc);
  }
  float* kvb = kvout + (size_t)b * CD * CD;
#pragma unroll
  for (int j = 0; j < 8; ++j) {
    const int c = blockIdx.x * 32 + mi * 16 + j + lhalf * 8;
    kvb[(size_t)c * CD + dcol] = acc[j];
  }
}

// ---------------------------------------------------------------------------
// Kernel 4: Q = relu(t@Wq+bq); num = Q@kv; denom = EPS + Q.(ksum+EPS);
//           out[b][d][n] = num[n][d] / denom[n]
// grid (NT/32, BN), block 256. One 64 KB LDS tile, recycled 3x via register
// staging across barriers.
// ---------------------------------------------------------------------------
__global__ __launch_bounds__(256) void k_out(
    const float* __restrict__ x, const float* __restrict__ Wq,
    const float* __restrict__ bq, const float* __restrict__ kv,
    const float* __restrict__ ksum, float* __restrict__ out) {
  __shared__ float s_lds[32 * CD];  // 64 KB, holds t -> Q -> num
  const int b = blockIdx.y;
  const int m0 = blockIdx.x * 32;
  const int tid = threadIdx.x;
  const int wave = tid >> 5, lane = tid & 31;
  const int l15 = lane & 15, lhalf = lane >> 4;
  const int koff = lhalf * 2;

  // phase 0: t tile into LDS
  const float* xb = x + (size_t)b * CD * NT;
  for (int i = tid; i < 32 * CD; i += 256) {
    int m = i & 31, c = i >> 5;
    s_lds[qidx(m, c)] = xb[(size_t)c * NT + m0 + m];
  }
  __syncthreads();

  // phase 1: Q sub-tiles (2 x 32 = 64, eight per wave) -> registers
  v8f qacc[8];
#pragma unroll
  for (int s = 0; s < 8; ++s) {
    const int sub = wave + s * 8;
    const int mi = sub & 1, ni = sub >> 1;
    const int col = ni * 16 + l15;
    v8f acc = {};
#pragma unroll 4
    for (int k0 = 0; k0 < CD; k0 += 4) {
      v2f a, bm;
      a.x = s_lds[qidx(mi * 16 + l15, k0 + koff)];
      a.y = s_lds[qidx(mi * 16 + l15, k0 + koff + 1)];
      bm.x = Wq[(size_t)(k0 + koff) * CD + col];
      bm.y = Wq[(size_t)(k0 + koff + 1) * CD + col];
      acc = wmma_f32(a, bm, acc);
    }
    const float bias = bq[col];
#pragma unroll
    for (int j = 0; j < 8; ++j) acc[j] = fmaxf(acc[j] + bias, 0.0f);
    qacc[s] = acc;
  }
  __syncthreads();
  // overwrite LDS with Q
#pragma unroll
  for (int s = 0; s < 8; ++s) {
    const int sub = wave + s * 8;
    const int mi = sub & 1, ni = sub >> 1;
#pragma unroll
    for (int j = 0; j < 8; ++j)
      s_lds[qidx(mi * 16 + j + lhalf * 8, ni * 16 + l15)] = qacc[s][j];
  }
  __syncthreads();

  // denominator for this thread's token row (m = tid & 31 is loop-invariant
  // for the phase-3 store pattern below)
  const int myM = tid & 31;
  float denom = EPSF;
  const float* ks = ksum + b * CD;
  for (int c = 0; c < CD; ++c)
    denom += s_lds[qidx(myM, c)] * (ks[c] + EPSF);

  // phase 2: num = Q @ kv  -> registers
  const float* kvb = kv + (size_t)b * CD * CD;
  v8f oacc[8];
#pragma unroll
  for (int s = 0; s < 8; ++s) {
    const int sub = wave + s * 8;
    const int mi = sub & 1, ni = sub >> 1;
    const int col = ni * 16 + l15;
    v8f acc = {};
#pragma unroll 4
    for (int k0 = 0; k0 < CD; k0 += 4) {
      v2f a, bm;
      a.x = s_lds[qidx(mi * 16 + l15, k0 + koff)];
      a.y = s_lds[qidx(mi * 16 + l15, k0 + koff + 1)];
      bm.x = kvb[(size_t)(k0 + koff) * CD + col];
      bm.y = kvb[(size_t)(k0 + koff + 1) * CD + col];
      acc = wmma_f32(a, bm, acc);
    }
    oacc[s] = acc;
  }
  __syncthreads();
  // overwrite LDS with num
#pragma unroll
  for (int s = 0; s < 8; ++s) {
    const int sub = wave + s * 8;
    const int mi = sub & 1, ni = sub >> 1;
#pragma unroll
    for (int j = 0; j < 8; ++j)
      s_lds[qidx(mi * 16 + j + lhalf * 8, ni * 16 + l15)] = oacc[s][j];
  }
  __syncthreads();

  // phase 3: divide + transposed store (coalesced along n)
  float* ob = out + (size_t)b * CD * NT;
  const float inv = 1.0f / denom;
  for (int i = tid; i < 32 * CD; i += 256) {
    const int m = i & 31;  // == myM for every iteration
    const int d = i >> 5;
    ob[(size_t)d * NT + m0 + m] = s_lds[qidx(m, d)] * inv;
  }
}

// ---------------------------------------------------------------------------
extern "C" void kernel_launch(void* const* d_in, const int* in_sizes, int n_in,
                              void* d_out, int out_size, void* d_ws,
                              size_t ws_size, hipStream_t stream) {
  (void)in_sizes; (void)n_in; (void)out_size; (void)ws_size;
  const float* x  = (const float*)d_in[0];
  const float* Wk = (const float*)d_in[1];
  const float* bk = (const float*)d_in[2];
  const float* Wq = (const float*)d_in[3];
  const float* bq = (const float*)d_in[4];
  const float* Wv = (const float*)d_in[5];
  const float* bv = (const float*)d_in[6];
  float* out = (float*)d_out;

  float* Kws  = (float*)d_ws;                       // B*N*C
  float* Vws  = Kws + (size_t)BN * NT * CD;         // B*N*C
  float* kvws = Vws + (size_t)BN * NT * CD;         // B*C*C
  float* ksum = kvws + (size_t)BN * CD * CD;        // B*C

  hipMemsetAsync(ksum, 0, (size_t)BN * CD * sizeof(float), stream);

  k_proj_kv<<<dim3(NT / 32, BN), 256, 0, stream>>>(x, Wk, bk, Wv, bv, Kws, Vws);
  k_ksum<<<dim3(CD / 256, NT / 256, BN), 256, 0, stream>>>(Kws, ksum);
  k_kvmat<<<dim3(CD / 32, CD / 64, BN), 256, 0, stream>>>(Kws, Vws, kvws);
  k_out<<<dim3(NT / 32, BN), 256, 0, stream>>>(x, Wq, bq, kvws, ksum, out);
}